// NeuralNetwork_75582834475335
// MI455X (gfx1250) — compile-verified
//
#include <hip/hip_runtime.h>
#include <math.h>

// ---------------- config ----------------
#define EPS    1e-8f
#define NBINS  4096          // histogram bins over dist in [-1, 3)
#define CAP    4096          // max candidates kept for final exact sort

typedef __attribute__((ext_vector_type(2))) float v2f;
typedef __attribute__((ext_vector_type(8))) float v8f;

// ---------------- kernel 0: init (zero hist/counters, compute xn = x/||x||) ----
__global__ void k_init(const float* __restrict__ x, float* __restrict__ xn,
                       int* __restrict__ hist, int* __restrict__ counters, int D)
{
    __shared__ float red[256];
    int t = threadIdx.x;
    for (int b = t; b < NBINS; b += 256) hist[b] = 0;
    if (t < 2) counters[t] = 0;
    float v = (t < D) ? x[t] : 0.0f;
    red[t] = v * v;
    __syncthreads();
    for (int s = 128; s > 0; s >>= 1) {
        if (t < s) red[t] += red[t + s];
        __syncthreads();
    }
    float inv = rsqrtf(red[0] + EPS);
    if (t < D) xn[t] = v * inv;
}

// ---------------- kernel 1: distances + histogram (bandwidth-bound pass) ------
__global__ void k_dist(const float* __restrict__ obs, const float* __restrict__ xn,
                       float* __restrict__ dists, int* __restrict__ hist,
                       int N, int D)
{
    __shared__ __align__(16) float sxn[256];
    __shared__ int lh[NBINS];
    int t = threadIdx.x;
    for (int b = t; b < NBINS; b += blockDim.x) lh[b] = 0;
    if (t < D) sxn[t] = xn[t];
    __syncthreads();

    const int D4 = D >> 2;
    const float4* o4 = (const float4*)obs;
    const float4* q4 = (const float4*)sxn;
    int stride = gridDim.x * blockDim.x;
    for (int i = blockIdx.x * blockDim.x + t; i < N; i += stride) {
        const float4* row = o4 + (size_t)i * D4;
        // prime L2 for this thread's next row (global_prefetch_b8)
        if (i + stride < N)
            __builtin_prefetch((const void*)(o4 + (size_t)(i + stride) * D4), 0, 0);
        float dot = 0.0f, ss = 0.0f;
#pragma unroll 8
        for (int d = 0; d < D4; ++d) {
            float4 o = row[d];
            float4 q = q4[d];
            dot += o.x*q.x + o.y*q.y + o.z*q.z + o.w*q.w;
            ss  += o.x*o.x + o.y*o.y + o.z*o.z + o.w*o.w;
        }
        float dist = 1.0f - dot * rsqrtf(ss + EPS);
        dists[i] = dist;
        int bin = (int)floorf((dist + 1.0f) * (NBINS * 0.25f));
        bin = min(max(bin, 0), NBINS - 1);
        atomicAdd(&lh[bin], 1);
    }
    __syncthreads();
    for (int b = t; b < NBINS; b += blockDim.x)
        if (lh[b]) atomicAdd(&hist[b], lh[b]);
}

// ---------------- kernel 2: find threshold bin (cum count >= K) ---------------
__global__ void k_thresh(const int* __restrict__ hist, int* __restrict__ counters, int K)
{
    int cum = 0, T = NBINS - 1;
    for (int b = 0; b < NBINS; ++b) {
        cum += hist[b];
        if (cum >= K) { T = b; break; }
    }
    counters[1] = T;
}

// ---------------- kernel 3: collect candidates (bin <= T) ---------------------
__global__ void k_collect(const float* __restrict__ dists, int* __restrict__ counters,
                          float* __restrict__ cd, int* __restrict__ ci, int N)
{
    int T = counters[1];
    int stride = gridDim.x * blockDim.x;
    for (int i = blockIdx.x * blockDim.x + threadIdx.x; i < N; i += stride) {
        float dist = dists[i];
        int bin = (int)floorf((dist + 1.0f) * (NBINS * 0.25f));
        bin = min(max(bin, 0), NBINS - 1);
        if (bin <= T) {
            int p = atomicAdd(&counters[0], 1);
            if (p < CAP) { cd[p] = dist; ci[p] = i; }
        }
    }
}

// ---------------- WMMA f32 matvec: vout = [relu](W * vin + b) -----------------
// One wave per 16-row tile. A = W tile (16x4 per step); B has vin in column 0.
// A layout (ISA 7.12.2, 32-bit A 16x4): lanes 0-15 -> K=0,1 ; lanes 16-31 -> K=2,3.
// D column 0: lane 0 holds M=0..7 in c[0..7], lane 16 holds M=8..15.
// Masking done arithmetically (unconditional b64 loads * 0/1), clamped-row
// addressing keeps every access in bounds. Requires n_in % 4 == 0 (128, 212).
__device__ inline void matvec_wmma(const float* __restrict__ W, const float* __restrict__ bias,
                                   const float* __restrict__ vin, float* __restrict__ vout,
                                   int n_out, int n_in, bool do_relu)
{
    int lane   = threadIdx.x & 31;
    int wave   = threadIdx.x >> 5;
    int nwaves = blockDim.x >> 5;
    int m      = lane & 15;
    int khalf  = (lane >> 4) << 1;            // 0 or 2
    int ntiles = (n_out + 15) >> 4;
    float bmask = (m == 0) ? 1.0f : 0.0f;     // column 0 of B carries vin

    for (int tile = wave; tile < ntiles; tile += nwaves) {
        int i0  = tile << 4;
        int row = i0 + m;
        float amask = (row < n_out) ? 1.0f : 0.0f;
        int rowc = min(row, n_out - 1);       // clamped, always in-bounds
        const float* wrow = W + (size_t)rowc * n_in + khalf;
        v8f c = {0.f, 0.f, 0.f, 0.f, 0.f, 0.f, 0.f, 0.f};
        for (int k0 = 0; k0 < n_in; k0 += 4) {
            v2f a = *(const v2f*)(wrow + k0);          // global_load_b64
            a.x *= amask; a.y *= amask;
            v2f b = *(const v2f*)(vin + k0 + khalf);   // ds_load_b64
            b.x *= bmask; b.y *= bmask;
            c = __builtin_amdgcn_wmma_f32_16x16x4_f32(
                    /*neg_a=*/false, a, /*neg_b=*/false, b,
                    /*c_mod=*/(short)0, c, /*reuse_a=*/false, /*reuse_b=*/false);
        }
        if (m == 0) {
            int mb = (lane >> 4) << 3;        // 0 or 8
            for (int r = 0; r < 8; ++r) {
                int orow = i0 + mb + r;
                if (orow < n_out) {
                    float v = c[r] + bias[orow];
                    vout[orow] = do_relu ? fmaxf(v, 0.0f) : v;
                }
            }
        }
    }
}

__device__ inline bool key_less(float d1, int i1, float d2, int i2)
{
    return (d1 < d2) || (d1 == d2 && i1 < i2);
}

// ---------------- kernel 4: sort candidates, top-k, MLP (WMMA), output --------
__global__ void __launch_bounds__(1024) k_final(
    const float* __restrict__ cd, const int* __restrict__ ci,
    const int* __restrict__ counters, const float* __restrict__ acs,
    const float* __restrict__ w_in, const float* __restrict__ b_in,
    const float* __restrict__ w2,   const float* __restrict__ b2,
    const float* __restrict__ w3,   const float* __restrict__ b3,
    const float* __restrict__ w_out,const float* __restrict__ b_out,
    float* __restrict__ out, int H, int K, int DA)
{
    __shared__ float sd[CAP];
    __shared__ int   si[CAP];
    __shared__ __align__(16) float kdist[260];
    __shared__ int   kidx[260];
    __shared__ __align__(16) float ha[260];
    __shared__ __align__(16) float hb[260];
    __shared__ __align__(16) float sw[260];
    __shared__ float sred[2];

    int t = threadIdx.x;
    int cnt = counters[0];
    if (cnt > CAP) cnt = CAP;

    for (int i = t; i < CAP; i += blockDim.x) {
        sd[i] = (i < cnt) ? cd[i] : __builtin_inff();
        si[i] = (i < cnt) ? ci[i] : 0x7fffffff;
    }
    __syncthreads();

    // bitonic sort ascending by (dist, idx)
    for (int k = 2; k <= CAP; k <<= 1) {
        for (int j = k >> 1; j > 0; j >>= 1) {
            for (int i = t; i < CAP; i += blockDim.x) {
                int l = i ^ j;
                if (l > i) {
                    bool up = ((i & k) == 0);
                    float di = sd[i], dl = sd[l];
                    int   ii = si[i], il = si[l];
                    bool sw2 = up ? key_less(dl, il, di, ii)
                                  : key_less(di, ii, dl, il);
                    if (sw2) { sd[i] = dl; si[i] = il; sd[l] = di; si[l] = ii; }
                }
            }
            __syncthreads();
        }
    }

    // top-K ascending, zero-pad MLP buffers
    for (int i = t; i < 260; i += blockDim.x) { kdist[i] = 0.f; kidx[i] = 0; ha[i] = 0.f; hb[i] = 0.f; sw[i] = 0.f; }
    __syncthreads();
    if (t < K) { kdist[t] = sd[t]; kidx[t] = si[t]; }
    __syncthreads();

    // MLP: K -> H -> H -> H -> K (f32 WMMA matvecs)
    matvec_wmma(w_in, b_in, kdist, ha, H, K, true);
    __syncthreads();
    matvec_wmma(w2, b2, ha, hb, H, H, true);
    __syncthreads();
    matvec_wmma(w3, b3, hb, ha, H, H, true);
    __syncthreads();
    matvec_wmma(w_out, b_out, ha, sw, K, H, false);   // logits in sw[0..K)
    __syncthreads();

    // softmax over K
    if (t == 0) {
        float mx = -__builtin_inff();
        for (int i = 0; i < K; ++i) mx = fmaxf(mx, sw[i]);
        sred[0] = mx;
    }
    __syncthreads();
    if (t < K) sw[t] = expf(sw[t] - sred[0]);
    __syncthreads();
    if (t == 0) {
        float s = 0.f;
        for (int i = 0; i < K; ++i) s += sw[i];
        sred[1] = 1.0f / s;
    }
    __syncthreads();
    if (t < K) sw[t] *= sred[1];
    __syncthreads();

    // weighted action sum -> out[DA]
    if (t < DA) {
        float acc = 0.f;
        for (int kk = 0; kk < K; ++kk)
            acc += sw[kk] * acs[(size_t)kidx[kk] * DA + t];
        out[t] = acc;
    }
}

// ---------------- launch ------------------------------------------------------
extern "C" void kernel_launch(void* const* d_in, const int* in_sizes, int n_in,
                              void* d_out, int out_size, void* d_ws, size_t ws_size,
                              hipStream_t stream)
{
    const float* obs   = (const float*)d_in[0];
    const float* acs   = (const float*)d_in[1];
    const float* x     = (const float*)d_in[2];
    const float* w_in  = (const float*)d_in[3];
    const float* b_in  = (const float*)d_in[4];
    const float* w2    = (const float*)d_in[5];
    const float* b2    = (const float*)d_in[6];
    const float* w3    = (const float*)d_in[7];
    const float* b3    = (const float*)d_in[8];
    const float* w_out = (const float*)d_in[9];
    const float* b_out = (const float*)d_in[10];

    const int D  = in_sizes[2];            // 32
    const int N  = in_sizes[0] / D;        // 2,000,000
    const int H  = in_sizes[4];            // 212
    const int K  = in_sizes[10];           // 128
    const int DA = in_sizes[1] / N;        // 16

    // workspace carve (assumes ws_size >= ~8.1 MB)
    char*  wsb      = (char*)d_ws;
    float* xn       = (float*)(wsb);                    // 256 floats
    int*   hist     = (int*)  (wsb + 1024);             // NBINS ints
    int*   counters = (int*)  (wsb + 1024 + NBINS * 4); // [0]=cand count, [1]=T
    float* cand_d   = (float*)(wsb + 32768);            // CAP floats
    int*   cand_i   = (int*)  (wsb + 32768 + CAP * 4);  // CAP ints
    float* dists    = (float*)(wsb + 65536);            // N floats

    float* out = (float*)d_out;

    k_init<<<1, 256, 0, stream>>>(x, xn, hist, counters, D);
    k_dist<<<2048, 256, 0, stream>>>(obs, xn, dists, hist, N, D);
    k_thresh<<<1, 1, 0, stream>>>(hist, counters, K);
    k_collect<<<1024, 256, 0, stream>>>(dists, counters, cand_d, cand_i, N);
    k_final<<<1, 1024, 0, stream>>>(cand_d, cand_i, counters, acs,
                                    w_in, b_in, w2, b2, w3, b3, w_out, b_out,
                                    out, H, K, DA);
}